// SKA_20581483282807
// MI455X (gfx1250) — compile-verified
//
#include <hip/hip_runtime.h>

// SKA: per-pixel dynamic grouped 3x3 conv.
// x:   [B=16, C=128, H=56, W=56]        fp32
// w:   [B=16, Cw=16, 9, H=56, W=56]     fp32  (channel c uses plane c % 16)
// out: [B, C, H, W]                     fp32
//
// out[b,c,h,x] = sum_{i,j in 0..2} xpad[b,c,h+i-1,x+j-1] * w[b,c%16,3i+j,h,x]
//
// Roofline: 58 M MACs vs >=80 MB HBM traffic -> ~1.4 FLOP/B, HBM-bound
// (~3.4 us floor at 23.3 TB/s). Strategy: pure streaming with b128 accesses,
// 8x register reuse of the dynamic taps (staged via gfx1250 async
// global->LDS, ASYNCcnt-ordered, barrier-free), NT stores.

#define BB   16
#define CC   128
#define HH   56
#define WW   56
#define CWCH 16            // weight channels
#define GRP  (CC / CWCH)   // 8 channel replicas per weight plane
#define HW   (HH * WW)     // 3136
#define W4   (WW / 4)      // 14 float4 groups per row
#define NTHREADS 256

typedef float v4f __attribute__((ext_vector_type(4)));
typedef int   v4i __attribute__((ext_vector_type(4)));

// ---- gfx1250 async global->LDS builtins (probe-guarded; safe fallback).
#if defined(__has_builtin)
#if __has_builtin(__builtin_amdgcn_global_load_async_to_lds_b128) && \
    __has_builtin(__builtin_amdgcn_s_wait_asynccnt)
#define USE_ASYNC_LDS 1
#endif
#endif

#ifndef USE_ASYNC_LDS
#define USE_ASYNC_LDS 0
#endif

#if USE_ASYNC_LDS
// Diagnostic from the probe round showed: param0 = 'int4 __device__ *',
// i.e. non-const v4i in the global AS; param1 = v4i in the shared AS.
typedef __attribute__((address_space(1))) v4i gas_v4i;
typedef __attribute__((address_space(3))) v4i las_v4i;
#endif

__global__ __launch_bounds__(NTHREADS)
void ska_dynconv3x3_kernel(const float* __restrict__ x,
                           const float* __restrict__ w,
                           float* __restrict__ out) {
    const int tid = threadIdx.x;
    const int idx = blockIdx.x * NTHREADS + tid;      // < 200704
    // idx = (h * 256 + bcw) * 14 + w4  -> single divide-by-14 decode
    const int w4  = idx % W4;
    const int t   = idx / W4;                         // = h*256 + bcw
    const int bcw = t & 255;                          // b*16 + cw
    const int h   = t >> 8;
    const int cw  = bcw & 15;
    const int b   = bcw >> 4;
    const int wc  = w4 * 4;                           // first output column

    // ---- Stage the 9 per-pixel taps (v4f over 4 columns). Read once from
    //      HBM, reused for all 8 channel replicas.
    const float* wp = w + ((size_t)(b * CWCH + cw) * 9 * HH + h) * WW + wc;
    v4f wk[9];

#if USE_ASYNC_LDS
    // Per-lane async copy global->LDS into this thread's private slot.
    // No cross-lane sharing -> no barrier; ASYNCcnt orders copy vs. ds reads.
    __shared__ v4f wstage[NTHREADS][9];               // 36 KB
#pragma unroll
    for (int k = 0; k < 9; ++k)
        __builtin_amdgcn_global_load_async_to_lds_b128(
            (gas_v4i*)(wp + k * HW),                  // casts away const; AS1
            (las_v4i*)&wstage[tid][k],                // AS3
            /*offset=*/0, /*cpol=*/0);
#endif

    // ---- Boundary predicates (zero padding). ----
    const bool rowok0 = (h > 0);
    const bool rowok2 = (h < HH - 1);
    const bool hasL   = (wc > 0);
    const bool hasR   = (wc + 4 < WW);

    const size_t base0 = ((size_t)(b * CC + cw) * HH + h) * WW + wc; // g = 0

#if USE_ASYNC_LDS
    __builtin_amdgcn_s_wait_asynccnt(0);              // taps landed in LDS
#pragma unroll
    for (int k = 0; k < 9; ++k)
        wk[k] = wstage[tid][k];                       // ds_load_b128
#else
#pragma unroll
    for (int k = 0; k < 9; ++k)
        wk[k] = __builtin_nontemporal_load((const v4f*)(wp + k * HW));
#endif

#pragma unroll
    for (int g = 0; g < GRP; ++g) {
        const size_t xbase = base0 + (size_t)g * CWCH * HW;
        const float* xp    = x + xbase;

        // Prefetch next channel replica's center row (global_prefetch_b8).
        if (g + 1 < GRP)
            __builtin_prefetch(xp + (size_t)CWCH * HW, 0, 0);

        v4f acc = (v4f){0.f, 0.f, 0.f, 0.f};

#pragma unroll
        for (int i = 0; i < 3; ++i) {                 // kernel row, r = i-1
            const bool rok = (i == 0) ? rowok0 : (i == 2) ? rowok2 : true;
            v4f   m = (v4f){0.f, 0.f, 0.f, 0.f};
            float l = 0.f, r = 0.f;
            if (rok) {
                const float* xr = xp + (i - 1) * WW;  // 16B-aligned
                m = *(const v4f*)xr;                  // global_load_b128
                if (hasL) l = xr[-1];
                if (hasR) r = xr[4];
            }
            const v4f wa  = wk[i * 3 + 0];            // tap j=0 (shift -1)
            const v4f wb  = wk[i * 3 + 1];            // tap j=1 (shift  0)
            const v4f wcv = wk[i * 3 + 2];            // tap j=2 (shift +1)

            acc.x = fmaf(wa.x, l,   acc.x);
            acc.y = fmaf(wa.y, m.x, acc.y);
            acc.z = fmaf(wa.z, m.y, acc.z);
            acc.w = fmaf(wa.w, m.z, acc.w);

            acc.x = fmaf(wb.x, m.x, acc.x);
            acc.y = fmaf(wb.y, m.y, acc.y);
            acc.z = fmaf(wb.z, m.z, acc.z);
            acc.w = fmaf(wb.w, m.w, acc.w);

            acc.x = fmaf(wcv.x, m.y, acc.x);
            acc.y = fmaf(wcv.y, m.z, acc.y);
            acc.z = fmaf(wcv.z, m.w, acc.z);
            acc.w = fmaf(wcv.w, r,   acc.w);
        }

        // Streaming result: non-temporal b128 store (TH=NT).
        __builtin_nontemporal_store(acc, (v4f*)(out + xbase));
    }
}

extern "C" void kernel_launch(void* const* d_in, const int* in_sizes, int n_in,
                              void* d_out, int out_size, void* d_ws, size_t ws_size,
                              hipStream_t stream) {
    const float* x = (const float*)d_in[0];   // [16,128,56,56]
    const float* w = (const float*)d_in[1];   // [16,16,9,56,56]
    float* out     = (float*)d_out;           // [16,128,56,56]

    const int total = BB * CWCH * HH * W4;    // 200704
    const int grid  = total / NTHREADS;       // 784 exactly
    ska_dynconv3x3_kernel<<<grid, NTHREADS, 0, stream>>>(x, w, out);
}